// WarpCTCLoss_50500225467027
// MI455X (gfx1250) — compile-verified
//
#include <hip/hip_runtime.h>
#include <stdint.h>

// Problem constants (match reference)
#define Bsz 64
#define Tn  1024
#define Vn  2048
#define Sn  256
#define Ln  513          // 2*S+1
#define LPAD 520         // padded row pitch (16B aligned)
#define CHUNK 8          // emit rows per TDM transfer
#define NEGF (-1e30f)

typedef float    f32x4 __attribute__((ext_vector_type(4)));
typedef uint32_t u32x4 __attribute__((ext_vector_type(4)));
typedef int      i32x4 __attribute__((ext_vector_type(4)));
typedef int      i32x8 __attribute__((ext_vector_type(8)));

// TDM builtin is an amdgcn-only builtin: it exists only in the device pass.
// Hard-error if the DEVICE toolchain lacks it; host pass gets a parse-only stub.
#if defined(__HIP_DEVICE_COMPILE__)
#  if !__has_builtin(__builtin_amdgcn_tensor_load_to_lds)
#    error "device toolchain lacks __builtin_amdgcn_tensor_load_to_lds"
#  endif
#  define CTC_TDM 1
#else
#  define CTC_TDM 0
#endif

// 1-D contiguous DMA of nelem fp32 from global -> LDS via Tensor Data Mover.
// Builds D# group0/group1 per CDNA5 ISA 8.3/8.4 (count=1, type=2, data_size=4B,
// 1-D tile of nelem elements; groups 2/3 zero => <=2D tensor).
__device__ __forceinline__ void tdm_load_1d(const void* lds_dst,
                                            const void* gsrc,
                                            uint32_t nelem) {
#if CTC_TDM
  uint64_t ga  = (uint64_t)(uintptr_t)gsrc;
  uint32_t lds = (uint32_t)(uintptr_t)lds_dst;   // low 32 bits of flat LDS addr == LDS offset
  u32x4 g0;
  g0[0] = 1u;                                        // count=1 valid descriptor, user mode
  g0[1] = lds;                                       // lds_addr [63:32]
  g0[2] = (uint32_t)ga;                              // global_addr[31:0]
  g0[3] = (uint32_t)((ga >> 32) & 0x01FFFFFFu)       // global_addr[56:32]
          | (2u << 30);                              // type=2 ("image")
  i32x8 g1;
  g1[0] = (int)(2u << 16);           // workgroup_mask=0, data_size=2 (4 bytes)
  g1[1] = (int)(nelem << 16);        // tensor_dim0 low16 (nelem < 65536)
  g1[2] = (int)(1u << 16);           // tensor_dim0 hi16=0, tensor_dim1 low16 = 1
  g1[3] = (int)(nelem << 16);        // tensor_dim1 hi16=0, tile_dim0 = nelem
  g1[4] = 0;                         // tile_dim1=0 (unused), tile_dim2=0
  g1[5] = (int)nelem;                // tensor_dim0_stride low32
  g1[6] = 0;
  g1[7] = 0;
  i32x4 z4 = {0, 0, 0, 0};
#if defined(__clang_major__) && (__clang_major__ >= 23)
  i32x8 z8 = {0, 0, 0, 0, 0, 0, 0, 0};
  __builtin_amdgcn_tensor_load_to_lds(g0, g1, z4, z4, z8, 0);
#else
  __builtin_amdgcn_tensor_load_to_lds(g0, g1, z4, z4, 0);
#endif
#else
  (void)lds_dst; (void)gsrc; (void)nelem;  // host parse-only stub
#endif
}

// ---------------------------------------------------------------------------
// Kernel A: per (b,t) row -> logsumexp + gather into emit[b,t,s] workspace.
// Streams logits exactly once (nontemporal), fully parallel over 65536 rows.
// ---------------------------------------------------------------------------
__global__ __launch_bounds__(256) void ctc_emit_kernel(
    const float* __restrict__ logits,
    const int*   __restrict__ targets,
    float*       __restrict__ emit) {
  const int t = blockIdx.x;
  const int b = blockIdx.y;
  const int tid = threadIdx.x;
  const int lane = tid & 31;
  const int wid = tid >> 5;

  __shared__ float rowl[Vn];
  __shared__ float red[8];

  const f32x4* rv = (const f32x4*)(logits + ((size_t)b * Tn + t) * Vn);
  // 8 floats per thread, nontemporal (single-use stream; keep L2 for emit)
  f32x4 x0 = __builtin_nontemporal_load(rv + tid);
  f32x4 x1 = __builtin_nontemporal_load(rv + tid + 256);
  ((f32x4*)rowl)[tid] = x0;
  ((f32x4*)rowl)[tid + 256] = x1;

  // ---- block max ----
  float m = fmaxf(fmaxf(fmaxf(x0[0], x0[1]), fmaxf(x0[2], x0[3])),
                  fmaxf(fmaxf(x1[0], x1[1]), fmaxf(x1[2], x1[3])));
#pragma unroll
  for (int off = 16; off > 0; off >>= 1) m = fmaxf(m, __shfl_xor(m, off, 32));
  if (lane == 0) red[wid] = m;
  __syncthreads();
  float M = red[0];
#pragma unroll
  for (int i = 1; i < 8; ++i) M = fmaxf(M, red[i]);
  __syncthreads();

  // ---- block sum of exp ----
  float s = __expf(x0[0] - M) + __expf(x0[1] - M) + __expf(x0[2] - M) +
            __expf(x0[3] - M) + __expf(x1[0] - M) + __expf(x1[1] - M) +
            __expf(x1[2] - M) + __expf(x1[3] - M);
#pragma unroll
  for (int off = 16; off > 0; off >>= 1) s += __shfl_xor(s, off, 32);
  if (lane == 0) red[wid] = s;
  __syncthreads();
  float tot = red[0];
#pragma unroll
  for (int i = 1; i < 8; ++i) tot += red[i];
  const float lse = M + __logf(tot);

  // ---- gather at extended labels: ext[2i]=blank(0), ext[2i+1]=targets[i] ----
  float* erow = emit + ((size_t)b * Tn + t) * LPAD;
  for (int ss = tid; ss < Ln; ss += 256) {
    const int idx = (ss & 1) ? targets[b * Sn + (ss >> 1)] : 0;
    erow[ss] = rowl[idx] - lse;
  }
}

// ---------------------------------------------------------------------------
// Kernel B: alpha DP. One block per utterance, one lane per DP state.
// Emit rows DMA'd to LDS in 8-row chunks (TDM, double-buffered) so the
// next chunk's transfer overlaps the current chunk's lse3 recursion.
// ---------------------------------------------------------------------------
__global__ __launch_bounds__(544) void ctc_alpha_kernel(
    const float* __restrict__ emit,
    const int*   __restrict__ targets,
    const int*   __restrict__ loglen,
    const int*   __restrict__ tgtlen,
    float*       __restrict__ out) {
  const int b = blockIdx.x;
  const int tid = threadIdx.x;

  __shared__ float alpha[2][LPAD];
  __shared__ float ebuf[2][CHUNK * LPAD];
  __shared__ int   tgt[Sn];

  const int len = loglen[b];

  // stage this utterance's targets once (coalesced)
  if (tid < Sn) tgt[tid] = targets[b * Sn + tid];

  const float* ebase = emit + (size_t)b * Tn * LPAD;

  // Load chunk 0 (rows t=0..7) via TDM
  if (tid < 32) {
    tdm_load_1d(&ebuf[0][0], ebase, CHUNK * LPAD);
    __builtin_amdgcn_s_wait_tensorcnt(0);
  }
  __syncthreads();

  // can_skip[s] = s odd, s>=3, targets[(s-1)/2] != targets[(s-3)/2]
  bool cs = false;
  if ((tid & 1) && tid >= 3 && tid < Ln)
    cs = tgt[(tid - 1) >> 1] != tgt[(tid - 3) >> 1];

  // alpha0
  if (tid < Ln) {
    float a = NEGF;
    if (tid == 0) a = ebuf[0][0];
    else if (tid == 1) a = ebuf[0][1];
    alpha[0][tid] = a;
  }
  __syncthreads();

  int cur = 0;
  const int nchunk = Tn / CHUNK;
  for (int c = 0; c < nchunk; ++c) {
    const int t0 = c * CHUNK;
    if (t0 >= len) break;                    // alpha frozen past len (uniform)
    const int sel = c & 1;

    // Prefetch next chunk into the other buffer (overlaps DP below)
    const bool want_next = (c + 1 < nchunk) && ((c + 1) * CHUNK < len);
    if (tid < 32 && want_next)
      tdm_load_1d(&ebuf[sel ^ 1][0], ebase + (size_t)(c + 1) * CHUNK * LPAD,
                  CHUNK * LPAD);

    // DP over rows of this chunk
    for (int r = 0; r < CHUNK; ++r) {
      const int t = t0 + r;
      if (t == 0) continue;                  // row 0 consumed by init
      if (t >= len) break;                   // uniform
      if (tid < Ln) {
        const float a0 = alpha[cur][tid];
        const float a1 = (tid >= 1) ? alpha[cur][tid - 1] : NEGF;
        const float a2 = cs ? alpha[cur][tid - 2] : NEGF;
        const float m = fmaxf(a0, fmaxf(a1, a2));
        const float sum = __expf(a0 - m) + __expf(a1 - m) + __expf(a2 - m);
        alpha[cur ^ 1][tid] = m + __logf(sum) + ebuf[sel][r * LPAD + tid];
      }
      __syncthreads();
      cur ^= 1;
    }

    // Make sure the prefetched chunk has landed before next iteration reads it
    if (tid < 32) __builtin_amdgcn_s_wait_tensorcnt(0);
    __syncthreads();
  }

  if (tid == 0) {
    const int i2 = 2 * tgtlen[b];
    const float x = alpha[cur][i2 - 1];
    const float y = alpha[cur][i2];
    const float m = fmaxf(x, y);
    out[b] = -(m + __logf(__expf(x - m) + __expf(y - m)));
  }
}

extern "C" void kernel_launch(void* const* d_in, const int* in_sizes, int n_in,
                              void* d_out, int out_size, void* d_ws, size_t ws_size,
                              hipStream_t stream) {
  const float* logits  = (const float*)d_in[0];   // [B,T,V] f32
  const int*   targets = (const int*)d_in[1];     // [B,S] i32
  const int*   loglen  = (const int*)d_in[2];     // [B] i32
  const int*   tgtlen  = (const int*)d_in[3];     // [B] i32
  float*       out     = (float*)d_out;           // [B] f32
  float*       emit    = (float*)d_ws;            // [B,T,LPAD] f32 scratch (~130 MiB)

  dim3 gridA(Tn, Bsz);
  ctc_emit_kernel<<<gridA, 256, 0, stream>>>(logits, targets, emit);
  ctc_alpha_kernel<<<Bsz, 544, 0, stream>>>(emit, targets, loglen, tgtlen, out);
}